// SimpleGAT_53197464928386
// MI455X (gfx1250) — compile-verified
//
#include <hip/hip_runtime.h>
#include <hip/hip_bf16.h>

#define NNODES 50000
#define NEDGES 400000
#define NEP    (NNODES + NEDGES)   /* 450000 edges incl. self loops */
#define HEADS  8
#define NEG_SLOPE 0.2f

typedef float v2f __attribute__((ext_vector_type(2)));
typedef float v8f __attribute__((ext_vector_type(8)));

// ---------------- fp32 WMMA GEMM: XP[N,M] = X[N,K] @ W[K,M] ----------------
// One wave per 16x16 output tile, V_WMMA_F32_16X16X4_F32, K stepped by 4.
// A-frag (16x4 f32, 2 VGPR): lanes 0-15 hold K={0,1}, lanes 16-31 hold K={2,3},
// row = lane&15.  B-frag mirrored (row-striped across lanes).  D: VGPR i holds
// row i (lanes 0-15) / row 8+i (lanes 16-31), col = lane&15.
__global__ void gemm_wmma_f32(const float* __restrict__ X,
                              const float* __restrict__ W,
                              float* __restrict__ XP,
                              int nRows, int K, int M) {
  const int wavesPerBlock = blockDim.x >> 5;
  int wave = blockIdx.x * wavesPerBlock + ((int)threadIdx.x >> 5);
  int colTiles = M >> 4;
  int tiles = (nRows >> 4) * colTiles;
  if (wave >= tiles) return;                 // uniform per wave: EXEC stays full
  int rowTile = wave / colTiles;
  int colTile = wave - rowTile * colTiles;
  int lane = threadIdx.x & 31;
  int r  = lane & 15;
  int kh = lane >> 4;                        // 0/1 -> K offsets {0,1}/{2,3}
  const float* arow = X + (size_t)(rowTile * 16 + r) * K;
  const float* bcol = W + colTile * 16 + r;
  v8f acc = {0.f, 0.f, 0.f, 0.f, 0.f, 0.f, 0.f, 0.f};
  for (int k = 0; k < K; k += 4) {
    int ka = k + 2 * kh;
    v2f a; a.x = arow[ka];               a.y = arow[ka + 1];
    v2f b; b.x = bcol[(size_t)ka * M];   b.y = bcol[(size_t)(ka + 1) * M];
    acc = __builtin_amdgcn_wmma_f32_16x16x4_f32(false, a, false, b,
                                                (short)0, acc, false, false);
  }
  int col = colTile * 16 + r;
  float* out = XP + (size_t)(rowTile * 16 + 8 * kh) * M + col;
#pragma unroll
  for (int i = 0; i < 8; ++i) out[(size_t)i * M] = acc[i];
}

// ---------------- per-(node,head) attention dot products -------------------
__global__ void att_dots(const float* __restrict__ XP,
                         const float* __restrict__ att_s,
                         const float* __restrict__ att_d,
                         float* __restrict__ a_s, float* __restrict__ a_d,
                         int C) {
  int t = blockIdx.x * blockDim.x + threadIdx.x;   // t = n*HEADS + h
  if (t >= NNODES * HEADS) return;
  int h = t & (HEADS - 1);
  const float* xp = XP + (size_t)t * C;            // xp[n,h,c] contiguous
  const float* as = att_s + h * C;
  const float* ad = att_d + h * C;
  float ss = 0.f, dd = 0.f;
  for (int c = 0; c < C; ++c) { float v = xp[c]; ss += v * as[c]; dd += v * ad[c]; }
  a_s[t] = ss; a_d[t] = dd;
}

// ---------------- edge helpers ---------------------------------------------
__device__ __forceinline__ void edge_sd(const int* __restrict__ ei, int i,
                                        int& s, int& d) {
  if (i < NEDGES) { s = ei[i]; d = ei[NEDGES + i]; }
  else            { s = i - NEDGES; d = s; }       // appended self loops
}
// monotone float<->uint mapping so atomicMax(u32) == float max
__device__ __forceinline__ unsigned enc_f32(float f) {
  unsigned u = __float_as_uint(f);
  return (u & 0x80000000u) ? ~u : (u | 0x80000000u);
}
__device__ __forceinline__ float dec_f32(unsigned u) {
  return (u & 0x80000000u) ? __uint_as_float(u & 0x7FFFFFFFu)
                           : __uint_as_float(~u);
}
__device__ __forceinline__ float lrelu(float e) {
  return (e > 0.f) ? e : NEG_SLOPE * e;
}

__global__ void zero_f32(float* __restrict__ p, long long n) {
  long long t = (long long)blockIdx.x * blockDim.x + threadIdx.x;
  if (t < n) p[t] = 0.f;
}

// pass 1: segment max of leaky_relu(a_s[src]+a_d[dst]) per (dst, head)
__global__ void edge_max(const int* __restrict__ ei,
                         const float* __restrict__ a_s,
                         const float* __restrict__ a_d,
                         unsigned* __restrict__ m_enc) {
  int t = blockIdx.x * blockDim.x + threadIdx.x;
  if (t >= NEP * HEADS) return;
  int i = t >> 3, h = t & (HEADS - 1);
  int s, d; edge_sd(ei, i, s, d);
  float e = lrelu(a_s[s * HEADS + h] + a_d[d * HEADS + h]);
  atomicMax(&m_enc[d * HEADS + h], enc_f32(e));
}

// pass 2: ex = exp(e - m[dst]); segment sum
__global__ void edge_exp(const int* __restrict__ ei,
                         const float* __restrict__ a_s,
                         const float* __restrict__ a_d,
                         const unsigned* __restrict__ m_enc,
                         float* __restrict__ exbuf,
                         float* __restrict__ ssum) {
  int t = blockIdx.x * blockDim.x + threadIdx.x;
  if (t >= NEP * HEADS) return;
  int i = t >> 3, h = t & (HEADS - 1);
  int s, d; edge_sd(ei, i, s, d);
  float e  = lrelu(a_s[s * HEADS + h] + a_d[d * HEADS + h]);
  float ex = expf(e - dec_f32(m_enc[d * HEADS + h]));
  exbuf[t] = ex;
  atomicAdd(&ssum[d * HEADS + h], ex);
}

// pass 3: alpha = ex / sum[dst]   (in place)
__global__ void edge_norm(const int* __restrict__ ei,
                          const float* __restrict__ ssum,
                          float* __restrict__ exbuf) {
  int t = blockIdx.x * blockDim.x + threadIdx.x;
  if (t >= NEP * HEADS) return;
  int i = t >> 3, h = t & (HEADS - 1);
  int s, d; edge_sd(ei, i, s, d);
  exbuf[t] = exbuf[t] / ssum[d * HEADS + h];
}

// pass 4: agg[dst, h, c] += xp[src, h, c] * alpha[e, h]
// one thread per float4 feature group -> b128 loads, amortized index math
__global__ void edge_agg4(const int* __restrict__ ei,
                          const float* __restrict__ XP,
                          const float* __restrict__ alpha,
                          float* __restrict__ agg, int C) {
  int HC  = HEADS * C;
  int HC4 = HC >> 2;                         // float4 groups per edge
  long long t = (long long)blockIdx.x * blockDim.x + threadIdx.x;
  if (t >= (long long)NEP * HC4) return;
  int i = (int)(t / HC4);
  int g = (int)(t - (long long)i * HC4);
  int f = g << 2;                            // C % 4 == 0 -> same head for all 4
  int h = f / C;
  int s, d; edge_sd(ei, i, s, d);
  float a = alpha[i * HEADS + h];
  const float4 v = *(const float4*)(XP + (size_t)s * HC + f);
  float* out = agg + (size_t)d * HC + f;
  atomicAdd(out + 0, v.x * a);
  atomicAdd(out + 1, v.y * a);
  atomicAdd(out + 2, v.z * a);
  atomicAdd(out + 3, v.w * a);
}

// head mean + bias + relu; write contiguous layer buffer and [N,80] slice
__global__ void finalize(const float* __restrict__ agg,
                         const float* __restrict__ bias,
                         float* __restrict__ hout, float* __restrict__ dout,
                         int C, int outOff) {
  int t = blockIdx.x * blockDim.x + threadIdx.x;   // n*C + c
  if (t >= NNODES * C) return;
  int n = t / C, c = t - n * C;
  const float* a = agg + (size_t)n * HEADS * C + c;
  float s = 0.f;
#pragma unroll
  for (int h = 0; h < HEADS; ++h) s += a[h * C];
  s = s * (1.0f / HEADS) + bias[c];
  s = fmaxf(s, 0.f);
  hout[t] = s;
  dout[(size_t)n * 80 + outOff + c] = s;
}

extern "C" void kernel_launch(void* const* d_in, const int* in_sizes, int n_in,
                              void* d_out, int out_size, void* d_ws, size_t ws_size,
                              hipStream_t stream) {
  const float* x  = (const float*)d_in[0];
  const int*   ei = (const int*)d_in[1];
  const float* Wl[3]  = {(const float*)d_in[2],  (const float*)d_in[6],  (const float*)d_in[10]};
  const float* asl[3] = {(const float*)d_in[3],  (const float*)d_in[7],  (const float*)d_in[11]};
  const float* adl[3] = {(const float*)d_in[4],  (const float*)d_in[8],  (const float*)d_in[12]};
  const float* bl[3]  = {(const float*)d_in[5],  (const float*)d_in[9],  (const float*)d_in[13]};

  // workspace layout (f32 elements)
  float* ws = (float*)d_ws;
  size_t o = 0;
  float*    xp    = ws + o; o += (size_t)NNODES * 256;     // projected feats (max M)
  float*    a_s   = ws + o; o += (size_t)NNODES * HEADS;
  float*    a_d   = ws + o; o += (size_t)NNODES * HEADS;
  unsigned* m_enc = (unsigned*)(ws + o); o += (size_t)NNODES * HEADS;
  float*    ssum  = ws + o; o += (size_t)NNODES * HEADS;
  float*    exbuf = ws + o; o += (size_t)NEP * HEADS;
  float*    agg   = ws + o; o += (size_t)NNODES * 256;
  float*    h1    = ws + o; o += (size_t)NNODES * 32;
  float*    h2    = ws + o; o += (size_t)NNODES * 16;
  float*    h3    = ws + o; o += (size_t)NNODES * 32;

  const int Cs[3]   = {32, 16, 32};
  const int Ks[3]   = {128, 32, 16};
  const int offs[3] = {0, 32, 48};
  float* hb[3] = {h1, h2, h3};

  const float* A = x;
  const int nhThreads = NNODES * HEADS;
  const int ehThreads = NEP * HEADS;

  for (int l = 0; l < 3; ++l) {
    const int C = Cs[l], K = Ks[l], M = HEADS * C;

    // zero segment-max (enc 0 == "below any real"), segment-sum, aggregation
    zero_f32<<<(nhThreads + 255) / 256, 256, 0, stream>>>((float*)m_enc, nhThreads);
    zero_f32<<<(nhThreads + 255) / 256, 256, 0, stream>>>(ssum, nhThreads);
    long long aggN = (long long)NNODES * M;
    zero_f32<<<(int)((aggN + 255) / 256), 256, 0, stream>>>(agg, aggN);

    // dense projection (fp32 WMMA)
    int tiles = (NNODES / 16) * (M / 16);
    gemm_wmma_f32<<<(tiles + 7) / 8, 256, 0, stream>>>(A, Wl[l], xp, NNODES, K, M);

    // per-(node,head) attention dots
    att_dots<<<(nhThreads + 255) / 256, 256, 0, stream>>>(xp, asl[l], adl[l], a_s, a_d, C);

    // edge softmax over incoming edges per (dst,head)
    edge_max <<<(ehThreads + 255) / 256, 256, 0, stream>>>(ei, a_s, a_d, m_enc);
    edge_exp <<<(ehThreads + 255) / 256, 256, 0, stream>>>(ei, a_s, a_d, m_enc, exbuf, ssum);
    edge_norm<<<(ehThreads + 255) / 256, 256, 0, stream>>>(ei, ssum, exbuf);

    // weighted scatter-aggregate (float4 per thread)
    long long aggT = (long long)NEP * (M >> 2);
    edge_agg4<<<(int)((aggT + 255) / 256), 256, 0, stream>>>(ei, xp, exbuf, agg, C);

    // head mean + bias + relu -> layer buffer + output slice
    finalize<<<(NNODES * C + 255) / 256, 256, 0, stream>>>(agg, bl[l], hb[l],
                                                           (float*)d_out, C, offs[l]);
    A = hb[l];
  }
}